// MovieRecGNN_34076270526866
// MI455X (gfx1250) — compile-verified
//
#include <hip/hip_runtime.h>

// ---------------------------------------------------------------------------
// MovieRecGNN on MI455X (gfx1250, wave32).
//
// Scatter raw features per edge (f32 atomics into L2-resident accumulators),
// then node-space GEMMs via V_WMMA_F32_16X16X4_F32 with a paired-row LDS
// weight layout so every B fragment is a single aligned ds_load_b64.
// ---------------------------------------------------------------------------

typedef __attribute__((ext_vector_type(2))) float v2f;
typedef __attribute__((ext_vector_type(8))) float v8f;

#define N_NODES 200000          // users == movies == entity rows touched
#define CDIM    64
#define WSZ     (CDIM * CDIM)   // 4096 floats per weight matrix
#define NTILES  (N_NODES / 16)  // 12500, exact

__device__ __forceinline__ v8f wmma4(v2f a, v2f b, v8f c) {
  // D = A(16x4,f32) x B(4x16,f32) + C(16x16,f32)
  return __builtin_amdgcn_wmma_f32_16x16x4_f32(
      /*neg_a=*/false, a, /*neg_b=*/false, b,
      /*c_mod=*/(short)0, c, /*reuse_a=*/false, /*reuse_b=*/false);
}

// Paired-row LDS layout: lds[(k>>1)*128 + 2n + (k&1)] = W[k][n]
// -> B fragment (W[ka][n], W[ka+1][n]) for even ka is one aligned b64.
__device__ __forceinline__ void stage_weight(float* lds, const float* W,
                                             int tid, int nthreads) {
  for (int i = tid; i < WSZ; i += nthreads) {
    int k = i >> 6, n = i & 63;
    lds[((k >> 1) << 7) + (n << 1) + (k & 1)] = W[i];
  }
}

__device__ __forceinline__ v2f ldsB(const float* lds, int pr, int n) {
  return *reinterpret_cast<const v2f*>(lds + (pr << 7) + (n << 1));
}

// ---------------------------------------------------------------------------
// Scatter: one wave per edge. agg[dst] += x[src] (64 f32), cnt[dst] += 1.
// Edge id is wave-uniform (readfirstlane) so src/dst indices come in via
// scalar loads; per-lane b64 gathers + coalesced f32 atomics into L2
// (3 x 51MB accumulators fit in the 192MB L2).
// ---------------------------------------------------------------------------
__global__ __launch_bounds__(256) void scatter_kernel(
    const int* __restrict__ esrc, const int* __restrict__ edst,
    const float* __restrict__ x, float* __restrict__ agg,
    float* __restrict__ cnt, int nE)
{
  int lane = threadIdx.x & 31;
  int wave = threadIdx.x >> 5;
  int e = blockIdx.x * (blockDim.x >> 5) + wave;
  if (e >= nE) return;
  e = __builtin_amdgcn_readfirstlane(e);   // provably uniform -> s_load indices
  int s = esrc[e];
  int d = edst[e];
  int c = lane << 1;
  v2f v = *reinterpret_cast<const v2f*>(x + (size_t)s * CDIM + c);
  atomicAdd(&agg[(size_t)d * CDIM + c], v.x);
  atomicAdd(&agg[(size_t)d * CDIM + c + 1], v.y);
  if (lane == 0) atomicAdd(&cnt[d], 1.0f);
}

// ---------------------------------------------------------------------------
// User side: msg = agg_u@W1 + (user_x*agg_u)@W2 + cnt_u*(b1+b2)
// out row = [user_x | msg]  (128 wide). One wave per 16-row tile.
// ---------------------------------------------------------------------------
__global__ __launch_bounds__(256) void user_gemm_kernel(
    const float* __restrict__ user_x, const float* __restrict__ agg_u,
    const float* __restrict__ cnt_u,
    const float* __restrict__ W1, const float* __restrict__ b1,
    const float* __restrict__ W2, const float* __restrict__ b2,
    float* __restrict__ out, int nTiles)
{
  __shared__ float lw[2 * WSZ];   // 32 KB
  stage_weight(lw, W1, threadIdx.x, blockDim.x);
  stage_weight(lw + WSZ, W2, threadIdx.x, blockDim.x);
  __syncthreads();

  int wave = threadIdx.x >> 5;
  int lane = threadIdx.x & 31;
  int tile = blockIdx.x * (blockDim.x >> 5) + wave;
  if (tile >= nTiles) return;            // wave-uniform: EXEC stays all-ones

  int base  = tile * 16;
  int m     = base + (lane & 15);        // A-matrix row for this lane
  int khalf = lane >> 4;                 // 0 (lanes 0-15) / 1 (lanes 16-31)
  int nlo   = lane & 15;                 // B/D column within N-tile

  const float* auRow = agg_u  + (size_t)m * CDIM;
  const float* uxRow = user_x + (size_t)m * CDIM;

  v8f acc[4] = {{}, {}, {}, {}};

#pragma unroll
  for (int k0 = 0; k0 < CDIM; k0 += 4) {
    int ka = k0 + (khalf << 1);          // even
    int pr = (k0 >> 1) + khalf;          // paired-row index in LDS
    v2f aA = *reinterpret_cast<const v2f*>(auRow + ka);
    v2f mx = *reinterpret_cast<const v2f*>(uxRow + ka);
    v2f aP = mx * aA;                    // v_pk_mul_f32
#pragma unroll
    for (int nt = 0; nt < 4; nt++) {
      int n = nt * 16 + nlo;
      acc[nt] = wmma4(aA, ldsB(lw, pr, n), acc[nt]);
      acc[nt] = wmma4(aP, ldsB(lw + WSZ, pr, n), acc[nt]);
    }
  }

  // D row r maps to base + r + 8*khalf; col = nt*16 + nlo.
  int rbase = base + (khalf << 3);
  float cnts[8];
#pragma unroll
  for (int r = 0; r < 8; r++) cnts[r] = cnt_u[rbase + r];

#pragma unroll
  for (int nt = 0; nt < 4; nt++) {
    int n = nt * 16 + nlo;
    float bias = b1[n] + b2[n];
#pragma unroll
    for (int r = 0; r < 8; r++) {
      int row = rbase + r;
      out[(size_t)row * 128 + 64 + n] = acc[nt][r] + cnts[r] * bias;
      out[(size_t)row * 128 + n]      = user_x[(size_t)row * CDIM + n];
    }
  }
}

// ---------------------------------------------------------------------------
// Movie side (4 fused matmul terms):
//   msg = agg_m@W1 + (movie_x*agg_m)@W2 + cnt_m*(b1+b2)          [NGCF]
//       + (agg_e/max(cnt_e,1))@Wrel + entity_x@Wroot + brgcn     [RGCN rows<200k]
// out row = [movie_x | msg]
// ---------------------------------------------------------------------------
__global__ __launch_bounds__(256) void movie_gemm_kernel(
    const float* __restrict__ movie_x, const float* __restrict__ entity_x,
    const float* __restrict__ agg_m, const float* __restrict__ cnt_m,
    const float* __restrict__ agg_e, const float* __restrict__ cnt_e,
    const float* __restrict__ W1, const float* __restrict__ b1,
    const float* __restrict__ W2, const float* __restrict__ b2,
    const float* __restrict__ Wrel, const float* __restrict__ Wroot,
    const float* __restrict__ brgcn, float* __restrict__ out, int nTiles)
{
  __shared__ float lw[4 * WSZ];   // 64 KB (<< 320 KB/WGP)
  stage_weight(lw,           W1,    threadIdx.x, blockDim.x);
  stage_weight(lw + 1 * WSZ, W2,    threadIdx.x, blockDim.x);
  stage_weight(lw + 2 * WSZ, Wrel,  threadIdx.x, blockDim.x);
  stage_weight(lw + 3 * WSZ, Wroot, threadIdx.x, blockDim.x);
  __syncthreads();

  int wave = threadIdx.x >> 5;
  int lane = threadIdx.x & 31;
  int tile = blockIdx.x * (blockDim.x >> 5) + wave;
  if (tile >= nTiles) return;            // wave-uniform

  int base  = tile * 16;
  int m     = base + (lane & 15);
  int khalf = lane >> 4;
  int nlo   = lane & 15;

  const float* amRow = agg_m    + (size_t)m * CDIM;
  const float* mxRow = movie_x  + (size_t)m * CDIM;
  const float* aeRow = agg_e    + (size_t)m * CDIM;
  const float* exRow = entity_x + (size_t)m * CDIM;
  float invE = 1.0f / fmaxf(cnt_e[m], 1.0f);

  v8f acc[4] = {{}, {}, {}, {}};

#pragma unroll
  for (int k0 = 0; k0 < CDIM; k0 += 4) {
    int ka = k0 + (khalf << 1);
    int pr = (k0 >> 1) + khalf;
    v2f aM = *reinterpret_cast<const v2f*>(amRow + ka);
    v2f mx = *reinterpret_cast<const v2f*>(mxRow + ka);
    v2f aP = mx * aM;                    // v_pk_mul_f32
    v2f ae = *reinterpret_cast<const v2f*>(aeRow + ka);
    v2f aE = ae * invE;                  // scalar splat mul
    v2f aX = *reinterpret_cast<const v2f*>(exRow + ka);
#pragma unroll
    for (int nt = 0; nt < 4; nt++) {
      int n = nt * 16 + nlo;
      acc[nt] = wmma4(aM, ldsB(lw,           pr, n), acc[nt]);
      acc[nt] = wmma4(aP, ldsB(lw + 1 * WSZ, pr, n), acc[nt]);
      acc[nt] = wmma4(aE, ldsB(lw + 2 * WSZ, pr, n), acc[nt]);
      acc[nt] = wmma4(aX, ldsB(lw + 3 * WSZ, pr, n), acc[nt]);
    }
  }

  int rbase = base + (khalf << 3);
  float cnts[8];
#pragma unroll
  for (int r = 0; r < 8; r++) cnts[r] = cnt_m[rbase + r];

#pragma unroll
  for (int nt = 0; nt < 4; nt++) {
    int n = nt * 16 + nlo;
    float bias12 = b1[n] + b2[n];
    float biasR  = brgcn[n];
#pragma unroll
    for (int r = 0; r < 8; r++) {
      int row = rbase + r;
      out[(size_t)row * 128 + 64 + n] = acc[nt][r] + cnts[r] * bias12 + biasR;
      out[(size_t)row * 128 + n]      = movie_x[(size_t)row * CDIM + n];
    }
  }
}

// ---------------------------------------------------------------------------
extern "C" void kernel_launch(void* const* d_in, const int* in_sizes, int n_in,
                              void* d_out, int out_size, void* d_ws, size_t ws_size,
                              hipStream_t stream) {
  const float* user_x   = (const float*)d_in[0];
  const float* movie_x  = (const float*)d_in[1];
  const float* entity_x = (const float*)d_in[2];
  const int*   um       = (const int*)d_in[3];
  const int*   me       = (const int*)d_in[4];
  const float* W1    = (const float*)d_in[5];
  const float* b1    = (const float*)d_in[6];
  const float* W2    = (const float*)d_in[7];
  const float* b2    = (const float*)d_in[8];
  const float* Wrel  = (const float*)d_in[9];
  const float* Wroot = (const float*)d_in[10];
  const float* brgcn = (const float*)d_in[11];

  const int nE_um = in_sizes[3] / 2;   // 1,000,000
  const int nE_me = in_sizes[4] / 2;   // 1,000,000

  // Workspace layout (floats):
  //   agg_m[200000*64] | agg_u[200000*64] | agg_e[200000*64] |
  //   cnt_m[200000] | cnt_u[200000] | cnt_e[200000]          == 156 MB
  float* ws    = (float*)d_ws;
  float* agg_m = ws;
  float* agg_u = ws + (size_t)1 * N_NODES * CDIM;
  float* agg_e = ws + (size_t)2 * N_NODES * CDIM;
  float* cnt_m = ws + (size_t)3 * N_NODES * CDIM;
  float* cnt_u = cnt_m + N_NODES;
  float* cnt_e = cnt_u + N_NODES;

  size_t zero_bytes = ((size_t)3 * N_NODES * CDIM + (size_t)3 * N_NODES) * sizeof(float);
  hipMemsetAsync(d_ws, 0, zero_bytes, stream);

  // --- scatter phase (one wave per edge) -----------------------------------
  {
    dim3 blk(256);                             // 8 edges per block
    dim3 grd_um((nE_um + 7) / 8);
    // movie NGCF: agg_m[dst] += movie_x[src]
    scatter_kernel<<<grd_um, blk, 0, stream>>>(um, um + nE_um, movie_x, agg_m, cnt_m, nE_um);
    // user NGCF over reversed edges: agg_u[src_orig] += user_x[dst_orig]
    scatter_kernel<<<grd_um, blk, 0, stream>>>(um + nE_um, um, user_x, agg_u, cnt_u, nE_um);
    // entity RGCN: agg_e[dst] += entity_x[src]
    dim3 grd_me((nE_me + 7) / 8);
    scatter_kernel<<<grd_me, blk, 0, stream>>>(me, me + nE_me, entity_x, agg_e, cnt_e, nE_me);
  }

  // --- GEMM + epilogue phase (WMMA) ---------------------------------------
  float* out_user  = (float*)d_out;
  float* out_movie = out_user + (size_t)N_NODES * 128;
  {
    dim3 blk(256);                       // 8 waves -> 8 row-tiles per block
    dim3 grd((NTILES + 7) / 8);
    user_gemm_kernel<<<grd, blk, 0, stream>>>(user_x, agg_u, cnt_u,
                                              W1, b1, W2, b2, out_user, NTILES);
    movie_gemm_kernel<<<grd, blk, 0, stream>>>(movie_x, entity_x, agg_m, cnt_m,
                                               agg_e, cnt_e, W1, b1, W2, b2,
                                               Wrel, Wroot, brgcn, out_movie, NTILES);
  }
}